// DynamicSparseAttention_52140902973956
// MI455X (gfx1250) — compile-verified
//
#include <hip/hip_runtime.h>
#include <math.h>

typedef __attribute__((ext_vector_type(2))) float v2f;
typedef __attribute__((ext_vector_type(8))) float v8f;

constexpr int B = 2, L = 2048, D = 512, H = 8, HD = 64;
constexpr int M = B * L;                 // 4096 rows

// ---------------- workspace layout (float offsets) ----------------
constexpr size_t OFF_Q    = 0;
constexpr size_t OFF_K    = OFF_Q    + (size_t)M * D;
constexpr size_t OFF_V    = OFF_K    + (size_t)M * D;
constexpr size_t OFF_GATE = OFF_V    + (size_t)M * D;
constexpr size_t OFF_CTX  = OFF_GATE + (size_t)M * D;
constexpr size_t OFF_MEAN = OFF_CTX  + (size_t)M * D;
constexpr size_t OFF_RSTD = OFF_MEAN + M;
constexpr size_t OFF_CM   = OFF_RSTD + M;        // colmean  B*D
constexpr size_t OFF_DVAR = OFF_CM   + (size_t)B * D;
constexpr size_t OFF_IMP  = OFF_DVAR + (size_t)B * D;
constexpr size_t OFF_TRR  = OFF_IMP  + M;
constexpr size_t OFF_DMEAN= OFF_TRR  + M;
constexpr size_t OFF_MAG  = OFF_DMEAN+ M;
constexpr size_t OFF_CHG  = OFF_MAG  + M;
constexpr size_t OFF_FS   = OFF_CHG  + M;
constexpr size_t OFF_SCAL = OFF_FS   + M;        // per-b scalars, stride 8
constexpr size_t OFF_LW   = OFF_SCAL + 64;       // int[M]
constexpr size_t OFF_FW   = OFF_LW   + M;        // int[M]
constexpr size_t OFF_KP   = OFF_FW   + M;        // uchar[M] (M/4 float slots)
constexpr size_t OFF_GM   = OFF_KP   + M / 4;    // uchar[M]
constexpr size_t OFF_ANY  = OFF_GM   + M / 4;    // int[B]

__device__ inline float wsum32(float v) {
  v += __shfl_xor(v, 16, 32); v += __shfl_xor(v, 8, 32);
  v += __shfl_xor(v, 4, 32);  v += __shfl_xor(v, 2, 32);
  v += __shfl_xor(v, 1, 32);
  return v;
}

__device__ inline v8f wmma4(v2f a, v2f b, v8f c) {
  return __builtin_amdgcn_wmma_f32_16x16x4_f32(false, a, false, b, (short)0, c,
                                               false, false);
}

// ---------------- init: zero control buffers ----------------
__global__ void init_kernel(float* ws) {
  int t = blockIdx.x * blockDim.x + threadIdx.x;
  if (t < M) ((unsigned char*)(ws + OFF_GM))[t] = 0;
  if (t < B) ((int*)(ws + OFF_ANY))[t] = 0;
}

// ---------------- column stats: colmean + per-d diff variance ----------------
__global__ void colstats_kernel(const float* __restrict__ x, float* ws) {
  int t = blockIdx.x * blockDim.x + threadIdx.x;
  if (t >= B * D) return;
  int b = t / D, d = t % D;
  const float* p = x + (size_t)b * L * D + d;
  float s = 0.f, s2 = 0.f, prev = 0.f, first = 0.f, lastv = 0.f;
  for (int i = 0; i < L; ++i) {
    float xv = p[(size_t)i * D];
    s += xv;
    if (i == 0) first = xv;
    else { float dd = xv - prev; s2 += dd * dd; }
    prev = xv; lastv = xv;
  }
  ws[OFF_CM + t] = s / (float)L;
  float n = (float)(L - 1);
  float s1 = lastv - first;                 // telescoped sum of diffs
  ws[OFF_DVAR + t] = (s2 - s1 * s1 / n) / (n - 1.f);   // ddof=1
}

// ---------------- freq[b] = mean_d dvar ----------------
__global__ void freq_kernel(float* ws) {
  __shared__ float buf[256];
  int b = blockIdx.x, t = threadIdx.x;
  float s = 0.f;
  for (int d = t; d < D; d += 256) s += ws[OFF_DVAR + (size_t)b * D + d];
  buf[t] = s; __syncthreads();
  for (int st = 128; st > 0; st >>= 1) { if (t < st) buf[t] += buf[t + st]; __syncthreads(); }
  if (t == 0) ws[OFF_SCAL + b * 8 + 5] = buf[0] / (float)D;
}

// ---------------- per-row statistics (one wave per row) ----------------
__global__ void __launch_bounds__(256) rowstats_kernel(const float* __restrict__ x,
                                                       float* ws) {
  int lane = threadIdx.x & 31;
  int row  = (blockIdx.x * blockDim.x + threadIdx.x) >> 5;
  if (row >= M) return;
  int b = row / L, i = row % L;
  const float* xr = x + (size_t)row * D;
  bool v1 = i + 1 < L, v2 = i + 2 < L, v3 = i + 3 < L, v4 = i + 4 < L, v5 = i + 5 < L;
  int lo = (i - 2 < 0) ? 0 : i - 2;
  int hi = (i + 2 > L - 1) ? L - 1 : i + 2;
  int cnt = hi - lo + 1;
  float a1=0,a2=0,a3=0,a4=0,a5=0,at=0,am=0,ax=0,afs=0;
  for (int d = lane; d < D; d += 32) {
    float x0 = xr[d];
    ax += x0; am += x0 * x0;
    if (v1) { float t1 = xr[D + d] - x0; a1 += fabsf(t1);
              at += fabsf(xr[D + d] - ws[OFF_CM + (size_t)b * D + d]); }
    if (v2) a2 += fabsf(xr[2 * D + d] - x0);
    if (v3) a3 += fabsf(xr[3 * D + d] - x0);
    if (v4) a4 += fabsf(xr[4 * D + d] - x0);
    if (v5) a5 += fabsf(xr[5 * D + d] - x0);
    // +-2 window stats
    float sm = 0.f;
    for (int p = lo; p <= hi; ++p) sm += x[((size_t)(b * L + p)) * D + d];
    float mean = sm / (float)cnt;
    float vv = 0.f;
    for (int p = lo; p <= hi; ++p) { float dd = x[((size_t)(b * L + p)) * D + d] - mean; vv += dd * dd; }
    afs += vv / (float)(cnt - 1);
  }
  a1 = wsum32(a1); a2 = wsum32(a2); a3 = wsum32(a3); a4 = wsum32(a4); a5 = wsum32(a5);
  at = wsum32(at); am = wsum32(am); ax = wsum32(ax); afs = wsum32(afs);
  if (lane == 0) {
    float invD = 1.f / (float)D;
    ws[OFF_IMP  + row] = (0.5f * a1 + 0.3f * (a2 * 0.5f) + 0.2f * (a4 * 0.25f)) * invD;
    ws[OFF_DMEAN+ row] = (0.4f * a1 + 0.3f * (a2 * 0.5f) + 0.2f * (a3 / 3.f) +
                          0.1f * (a5 * 0.2f)) * invD;
    ws[OFF_CHG + row] = v1 ? a1 * invD : 0.f;
    ws[OFF_TRR + row] = v1 ? at * invD : 0.f;
    ws[OFF_MAG + row] = sqrtf(am);
    float freq = ws[OFF_SCAL + b * 8 + 5];
    ws[OFF_FS + row] = (i == 0 || i == L - 1) ? 0.f : (afs * invD) / (freq + 1e-6f);
    float mu = ax * invD;
    float var = fmaxf(am * invD - mu * mu, 0.f);
    ws[OFF_MEAN + row] = mu;
    ws[OFF_RSTD + row] = rsqrtf(var + 1e-5f);
  }
}

// ---------------- per-b reductions over L ----------------
__global__ void rowreduce_kernel(float* ws) {
  int b = blockIdx.x, t = threadIdx.x;
  __shared__ float buf[256];
  __shared__ float sh_mean;
  float mn = 1e30f, mx = -1e30f;
  for (int i = t; i < L; i += 256) { float v = ws[OFF_IMP + (size_t)b * L + i]; mn = fminf(mn, v); mx = fmaxf(mx, v); }
  buf[t] = mn; __syncthreads();
  for (int s = 128; s > 0; s >>= 1) { if (t < s) buf[t] = fminf(buf[t], buf[t + s]); __syncthreads(); }
  if (t == 0) ws[OFF_SCAL + b * 8 + 0] = buf[0];
  __syncthreads();
  buf[t] = mx; __syncthreads();
  for (int s = 128; s > 0; s >>= 1) { if (t < s) buf[t] = fmaxf(buf[t], buf[t + s]); __syncthreads(); }
  if (t == 0) ws[OFF_SCAL + b * 8 + 1] = buf[0];
  __syncthreads();
  mn = 1e30f; mx = -1e30f;
  for (int i = t; i < L; i += 256) { float v = ws[OFF_TRR + (size_t)b * L + i]; mn = fminf(mn, v); mx = fmaxf(mx, v); }
  buf[t] = mn; __syncthreads();
  for (int s = 128; s > 0; s >>= 1) { if (t < s) buf[t] = fminf(buf[t], buf[t + s]); __syncthreads(); }
  if (t == 0) ws[OFF_SCAL + b * 8 + 2] = buf[0];
  __syncthreads();
  buf[t] = mx; __syncthreads();
  for (int s = 128; s > 0; s >>= 1) { if (t < s) buf[t] = fmaxf(buf[t], buf[t + s]); __syncthreads(); }
  if (t == 0) ws[OFF_SCAL + b * 8 + 3] = buf[0];
  __syncthreads();
  float s = 0.f;
  for (int i = t; i < L; i += 256) s += ws[OFF_DMEAN + (size_t)b * L + i];
  buf[t] = s; __syncthreads();
  for (int st = 128; st > 0; st >>= 1) { if (t < st) buf[t] += buf[t + st]; __syncthreads(); }
  if (t == 0) sh_mean = buf[0] / (float)L;
  __syncthreads();
  float ss = 0.f;
  for (int i = t; i < L; i += 256) { float d = ws[OFF_DMEAN + (size_t)b * L + i] - sh_mean; ss += d * d; }
  buf[t] = ss; __syncthreads();
  for (int st = 128; st > 0; st >>= 1) { if (t < st) buf[t] += buf[t + st]; __syncthreads(); }
  if (t == 0)
    ws[OFF_SCAL + b * 8 + 4] = sh_mean + 0.5f * sqrtf(buf[0] / (float)(L - 1)); // thr
}

// ---------------- windows (lw/fw) + keypoints ----------------
__global__ void windowkp_kernel(float* ws) {
  int t = blockIdx.x * blockDim.x + threadIdx.x;
  if (t >= M) return;
  int b = t / L, i = t % L;
  float mn = ws[OFF_SCAL + b * 8 + 0], mx = ws[OFF_SCAL + b * 8 + 1];
  float impn = (ws[OFF_IMP + t] - mn) / (mx - mn + 1e-6f);
  int lwv = (int)rintf(64.f * (0.5f + 0.5f * impn));
  lwv = lwv < 2 ? 2 : (lwv > 128 ? 128 : lwv);
  mn = ws[OFF_SCAL + b * 8 + 2]; mx = ws[OFF_SCAL + b * 8 + 3];
  float trn = (ws[OFF_TRR + t] - mn) / (mx - mn + 1e-6f);
  int fwv = (int)rintf(32.f * (0.5f + 0.5f * trn));
  fwv = fwv < 1 ? 1 : (fwv > 32 ? 32 : fwv);
  ((int*)(ws + OFF_LW))[t] = lwv;
  ((int*)(ws + OFF_FW))[t] = fwv;
  float thr = ws[OFF_SCAL + b * 8 + 4];
  float dmv = ws[OFF_DMEAN + t];
  bool kpv;
  if (i == 0 || i == L - 1) kpv = dmv > thr;
  else kpv = (dmv > ws[OFF_DMEAN + t - 1]) && (dmv > ws[OFF_DMEAN + t + 1]) && (dmv > thr);
  ((unsigned char*)(ws + OFF_KP))[t] = kpv ? 1 : 0;
  if (kpv) atomicOr(((int*)(ws + OFF_ANY)) + b, 1);
}

// ---------------- per-(b,segment) top-4 global indices ----------------
__global__ void topk_kernel(float* ws) {
  int w = threadIdx.x >> 5;       // 0..7 -> (b,seg)
  int lane = threadIdx.x & 31;
  int b = w >> 2, seg = w & 3;
  int base = seg * (L / 4);
  int ex0 = -1, ex1 = -1, ex2 = -1, ex3 = -1;
  for (int round = 0; round < 4; ++round) {
    float bestv = -1e30f; int besti = 1 << 30;
    for (int tt = 0; tt < (L / 4) / 32; ++tt) {
      int idx = base + lane + tt * 32;
      if (idx == ex0 || idx == ex1 || idx == ex2 || idx == ex3) continue;
      int row = b * L + idx;
      float v = 0.3f * ws[OFF_MAG + row] + 0.4f * ws[OFF_CHG + row] + 0.3f * ws[OFF_FS + row];
      if (v > bestv || (v == bestv && idx < besti)) { bestv = v; besti = idx; }
    }
    for (int m = 16; m >= 1; m >>= 1) {
      float ov = __shfl_xor(bestv, m, 32);
      int   oi = __shfl_xor(besti, m, 32);
      if (ov > bestv || (ov == bestv && oi < besti)) { bestv = ov; besti = oi; }
    }
    if (lane == 0) ((unsigned char*)(ws + OFF_GM))[b * L + besti] = 1;
    if (round == 0) ex0 = besti; else if (round == 1) ex1 = besti;
    else if (round == 2) ex2 = besti; else ex3 = besti;
  }
}

// ---------------- WMMA GEMM: OUT = A(X) @ W + bias ----------------
// AMODE 0: A = (X-mean)*rstd*gamma+beta | 1: A = X | 2: A = X * X2 (gate)
// OMODE 0: raw | 1: sigmoid
template <int AMODE, int OMODE>
__global__ void __launch_bounds__(256) gemm_kernel(
    const float* __restrict__ X, const float* __restrict__ W,
    const float* __restrict__ bias,
    const float* __restrict__ gamma, const float* __restrict__ beta,
    const float* __restrict__ mean, const float* __restrict__ rstd,
    const float* __restrict__ X2, float* __restrict__ OUT) {
  int lane = threadIdx.x & 31;
  int wid = (blockIdx.x * blockDim.x + threadIdx.x) >> 5;
  constexpr int NT = D / 16;
  int mt = wid / NT, nt = wid % NT;
  if (mt >= M / 16) return;
  int m0 = mt * 16, n0 = nt * 16;
  int half = lane >> 4, nl = lane & 15;
  int kbase = half * 2;
  int mA = m0 + nl;            // A-fragment row for this lane
  int n  = n0 + nl;            // output column for this lane
  float mu = 0.f, rsd = 0.f;
  if constexpr (AMODE == 0) { mu = mean[mA]; rsd = rstd[mA]; }
  const float* xrow  = X + (size_t)mA * D;
  const float* x2row = (AMODE == 2) ? (X2 + (size_t)mA * D) : nullptr;
  v8f acc = {};
  for (int k0 = 0; k0 < D; k0 += 4) {
    v2f a = *(const v2f*)(xrow + k0 + kbase);
    if constexpr (AMODE == 0) {
      v2f g  = *(const v2f*)(gamma + k0 + kbase);
      v2f bb = *(const v2f*)(beta + k0 + kbase);
      a[0] = (a[0] - mu) * rsd * g[0] + bb[0];
      a[1] = (a[1] - mu) * rsd * g[1] + bb[1];
    } else if constexpr (AMODE == 2) {
      v2f g2 = *(const v2f*)(x2row + k0 + kbase);
      a[0] *= g2[0]; a[1] *= g2[1];
    }
    v2f bf;
    bf[0] = W[(size_t)(k0 + kbase) * D + n];
    bf[1] = W[(size_t)(k0 + kbase + 1) * D + n];
    acc = wmma4(a, bf, acc);
  }
  float bn = bias[n];
#pragma unroll
  for (int r = 0; r < 8; ++r) {
    float v = acc[r] + bn;
    if constexpr (OMODE == 1) v = 1.f / (1.f + expf(-v));
    OUT[(size_t)(m0 + r + 8 * half) * D + n] = v;
  }
}

// ---------------- masked flash attention, one 16-row tile per wave ----------------
__global__ void __launch_bounds__(128) attn_kernel(
    const float* __restrict__ Q, const float* __restrict__ Km,
    const float* __restrict__ Vm, const int* __restrict__ lw,
    const int* __restrict__ fw, const unsigned char* __restrict__ kp,
    const unsigned char* __restrict__ gm, const int* __restrict__ anykp,
    float* __restrict__ CTX) {
  __shared__ __align__(16) float plds[4][16 * 16];
  int lane = threadIdx.x & 31;
  int w = threadIdx.x >> 5;
  int wid = blockIdx.x * 4 + w;
  constexpr int IT = L / 16;
  int b = wid / (H * IT);
  int rem = wid % (H * IT);
  int h = rem / IT;
  int it = rem % IT;
  int i0 = it * 16;
  int half = lane >> 4;
  int nl = lane & 15;
  int kbase = half * 2;

  // preload 16x64 Q tile as A fragments (16 k-steps of 4)
  v2f qreg[16];
  const float* qrow = Q + ((size_t)(b * L + i0 + nl)) * D + h * HD;
#pragma unroll
  for (int kk = 0; kk < 16; ++kk) qreg[kk] = *(const v2f*)(qrow + kk * 4 + kbase);

  int lwr[8], fwr[8];
  unsigned char kpr[8];
#pragma unroll
  for (int r = 0; r < 8; ++r) {
    int i = i0 + r + 8 * half;
    lwr[r] = lw[b * L + i]; fwr[r] = fw[b * L + i]; kpr[r] = kp[b * L + i];
  }
  bool any = anykp[b] != 0;

  float mrun[8], srun[8];
#pragma unroll
  for (int r = 0; r < 8; ++r) { mrun[r] = -1e30f; srun[r] = 0.f; }
  v8f acc[4];
#pragma unroll
  for (int t = 0; t < 4; ++t) acc[t] = (v8f){};

  for (int jt = 0; jt < L / 16; ++jt) {
    int j0 = jt * 16;
    int j = j0 + nl;
    bool gmj = gm[b * L + j] != 0;
    bool kpj = kp[b * L + j] != 0;
    bool fbj = (j == 0) || (j == 511) || (j == 1023) || (j == 1535) || (j == 2047);

    // S = Q K^T for this 16x16 tile
    v8f s = {};
    const float* krow = Km + ((size_t)(b * L + j0 + nl)) * D + h * HD;
#pragma unroll
    for (int kk = 0; kk < 16; ++kk) {
      v2f bf = *(const v2f*)(krow + kk * 4 + kbase);
      s = wmma4(qreg[kk], bf, s);
    }

    float p[8];
#pragma unroll
    for (int r = 0; r < 8; ++r) {
      int i = i0 + r + 8 * half;
      float val = s[r] * 0.125f;  // 1/sqrt(64)
      bool allow = ((j <= i) && (j >= i - lwr[r])) ||
                   ((j > i) && (j <= i + fwr[r])) ||
                   (any ? (kpr[r] || kpj) : fbj) || gmj;
      val = allow ? val : -1e9f;
      float rm = val;
      rm = fmaxf(rm, __shfl_xor(rm, 1, 32));
      rm = fmaxf(rm, __shfl_xor(rm, 2, 32));
      rm = fmaxf(rm, __shfl_xor(rm, 4, 32));
      rm = fmaxf(rm, __shfl_xor(rm, 8, 32));
      float mnew = fmaxf(mrun[r], rm);
      float sc = expf(mrun[r] - mnew);
      float pe = expf(val - mnew);
      float rs = pe;
      rs += __shfl_xor(rs, 1, 32); rs += __shfl_xor(rs, 2, 32);
      rs += __shfl_xor(rs, 4, 32); rs += __shfl_xor(rs, 8, 32);
      srun[r] = srun[r] * sc + rs;
      mrun[r] = mnew;
#pragma unroll
      for (int t = 0; t < 4; ++t) acc[t][r] *= sc;
      p[r] = pe;
    }

    // re-layout P: C/D layout -> A layout via LDS
    __syncthreads();
#pragma unroll
    for (int r = 0; r < 8; ++r) plds[w][(r + 8 * half) * 16 + nl] = p[r];
    __syncthreads();

    // O += P @ V_tile  (4 column tiles of 16)
#pragma unroll
    for (int kk = 0; kk < 4; ++kk) {
      v2f af = *(const v2f*)(&plds[w][nl * 16 + kk * 4 + kbase]);
#pragma unroll
      for (int t = 0; t < 4; ++t) {
        const float* vp = Vm + ((size_t)(b * L + j0 + kk * 4 + kbase)) * D + h * HD + t * 16 + nl;
        v2f bf;
        bf[0] = vp[0];
        bf[1] = vp[D];
        acc[t] = wmma4(af, bf, acc[t]);
      }
    }
  }

#pragma unroll
  for (int r = 0; r < 8; ++r) {
    float inv = 1.f / srun[r];
    int i = i0 + r + 8 * half;
    float* crow = CTX + ((size_t)(b * L + i)) * D + h * HD;
#pragma unroll
    for (int t = 0; t < 4; ++t) crow[t * 16 + nl] = acc[t][r] * inv;
  }
}

// ---------------- host launcher ----------------
extern "C" void kernel_launch(void* const* d_in, const int* in_sizes, int n_in,
                              void* d_out, int out_size, void* d_ws, size_t ws_size,
                              hipStream_t stream) {
  (void)in_sizes; (void)n_in; (void)out_size; (void)ws_size;
  const float* x  = (const float*)d_in[0];
  const float* Wq = (const float*)d_in[1];  const float* bq = (const float*)d_in[2];
  const float* Wk = (const float*)d_in[3];  const float* bk = (const float*)d_in[4];
  const float* Wv = (const float*)d_in[5];  const float* bv = (const float*)d_in[6];
  const float* Wo = (const float*)d_in[7];  const float* bo = (const float*)d_in[8];
  const float* Wg = (const float*)d_in[9];  const float* bg = (const float*)d_in[10];
  const float* qg = (const float*)d_in[11]; const float* qb = (const float*)d_in[12];
  const float* kg = (const float*)d_in[13]; const float* kb = (const float*)d_in[14];
  const float* vg = (const float*)d_in[15]; const float* vb = (const float*)d_in[16];
  float* ws = (float*)d_ws;
  float* out = (float*)d_out;

  init_kernel<<<16, 256, 0, stream>>>(ws);
  colstats_kernel<<<(B * D + 255) / 256, 256, 0, stream>>>(x, ws);
  freq_kernel<<<B, 256, 0, stream>>>(ws);
  rowstats_kernel<<<M / 8, 256, 0, stream>>>(x, ws);
  rowreduce_kernel<<<B, 256, 0, stream>>>(ws);
  windowkp_kernel<<<(M + 255) / 256, 256, 0, stream>>>(ws);
  topk_kernel<<<1, 256, 0, stream>>>(ws);

  int gblocks = (M / 16) * (D / 16) / 8;  // 8 waves per block
  gemm_kernel<0, 0><<<gblocks, 256, 0, stream>>>(x, Wq, bq, qg, qb,
      ws + OFF_MEAN, ws + OFF_RSTD, nullptr, ws + OFF_Q);
  gemm_kernel<0, 0><<<gblocks, 256, 0, stream>>>(x, Wk, bk, kg, kb,
      ws + OFF_MEAN, ws + OFF_RSTD, nullptr, ws + OFF_K);
  gemm_kernel<0, 0><<<gblocks, 256, 0, stream>>>(x, Wv, bv, vg, vb,
      ws + OFF_MEAN, ws + OFF_RSTD, nullptr, ws + OFF_V);
  gemm_kernel<1, 1><<<gblocks, 256, 0, stream>>>(x, Wg, bg, nullptr, nullptr,
      nullptr, nullptr, nullptr, ws + OFF_GATE);

  attn_kernel<<<(B * H * (L / 16)) / 4, 128, 0, stream>>>(
      ws + OFF_Q, ws + OFF_K, ws + OFF_V,
      (const int*)(ws + OFF_LW), (const int*)(ws + OFF_FW),
      (const unsigned char*)(ws + OFF_KP), (const unsigned char*)(ws + OFF_GM),
      (const int*)(ws + OFF_ANY), ws + OFF_CTX);

  gemm_kernel<2, 0><<<gblocks, 256, 0, stream>>>(ws + OFF_CTX, Wo, bo, nullptr,
      nullptr, nullptr, nullptr, ws + OFF_GATE, out);
}